// EdgeAwareFeatureExtractor_24481313587545
// MI455X (gfx1250) — compile-verified
//
#include <hip/hip_runtime.h>
#include <hip/hip_bf16.h>

// ---------------------------------------------------------------------------
// EdgeAwareFeatureExtractor (DRConv2d, k=1) for gfx1250 / MI455X.
// Strategy: route-then-GEMM. Guide argmax bins pixels per (batch,region);
// each bin runs an exact fp32 GEMM [pixels x 64] x [64 x 64] using chained
// V_WMMA_F32_16X16X4_F32 (16 K-steps), B-matrix staged in LDS.
// B=4, C=64, H=W=256, r=8. Problem constants are baked in.
// ---------------------------------------------------------------------------

#define B_     4
#define C_     64
#define HW_    65536          // 256*256
#define R_     8

typedef float v2f __attribute__((ext_vector_type(2)));
typedef float v8f __attribute__((ext_vector_type(8)));

// ---- K0: zero the 32 bin counters --------------------------------------
__global__ void k0_zero_counts(int* counts) {
    if (threadIdx.x < B_ * R_) counts[threadIdx.x] = 0;
}

// ---- K1: x1 = W1*[rgb;edge]+b1 (pixel-major), guide argmax, bin pixels --
__global__ void __launch_bounds__(256)
k1_x1_guide_bin(const float* __restrict__ rgb, const float* __restrict__ edge,
                const float* __restrict__ W1,  const float* __restrict__ b1,
                const float* __restrict__ Wg,  const float* __restrict__ bg,
                float* __restrict__ x1t, int* __restrict__ counts,
                int* __restrict__ binned) {
    int gid = blockIdx.x * 256 + threadIdx.x;      // [0, B*HW)
    int b = gid >> 16;
    int p = gid & (HW_ - 1);

    const float* rb = rgb + (size_t)b * 3 * HW_;
    float in0 = rb[p];
    float in1 = rb[HW_ + p];
    float in2 = rb[2 * HW_ + p];
    float in3 = edge[(size_t)b * HW_ + p];

    float ga[R_];
#pragma unroll
    for (int g = 0; g < R_; ++g) ga[g] = bg[g];    // uniform -> scalar loads

    float* xrow = x1t + (size_t)gid * C_;          // pixel-major row
#pragma unroll
    for (int c = 0; c < C_; ++c) {
        float v = b1[c] + W1[c * 4 + 0] * in0 + W1[c * 4 + 1] * in1
                        + W1[c * 4 + 2] * in2 + W1[c * 4 + 3] * in3;
        xrow[c] = v;
#pragma unroll
        for (int g = 0; g < R_; ++g) ga[g] += Wg[g * C_ + c] * v;
    }

    // argmax (first max wins, matching jnp.argmax)
    int gi = 0; float best = ga[0];
#pragma unroll
    for (int g = 1; g < R_; ++g) if (ga[g] > best) { best = ga[g]; gi = g; }

    int bin  = b * R_ + gi;
    int slot = atomicAdd(&counts[bin], 1);
    binned[(size_t)bin * HW_ + slot] = p;
}

// ---- K2: deterministic per-(b,c) mean over HW (fixed-order tree) --------
__global__ void __launch_bounds__(256)
k2_mean(const float* __restrict__ x1t, float* __restrict__ gmean) {
    __shared__ float red[256];
    int bc = blockIdx.x;                 // b*64 + c
    int b  = bc >> 6;
    int c  = bc & 63;
    float s = 0.0f;
    for (int p = threadIdx.x; p < HW_; p += 256)
        s += x1t[((size_t)b * HW_ + p) * C_ + c];
    red[threadIdx.x] = s;
    __syncthreads();
    for (int off = 128; off > 0; off >>= 1) {
        if (threadIdx.x < off) red[threadIdx.x] += red[threadIdx.x + off];
        __syncthreads();
    }
    if (threadIdx.x == 0) gmean[bc] = red[0] * (1.0f / (float)HW_);
}

// ---- K3: t = sigmoid(gmean @ Wk1^T + bk1), 256 outputs ------------------
__global__ void k3_sigmoid(const float* __restrict__ gmean,
                           const float* __restrict__ Wk1,
                           const float* __restrict__ bk1,
                           float* __restrict__ tbuf) {
    int tid = threadIdx.x;               // B * r*r = 256
    int b = tid >> 6, j = tid & 63;
    const float* gm = gmean + b * C_;
    float s = bk1[j];
#pragma unroll
    for (int c = 0; c < C_; ++c) s += Wk1[j * C_ + c] * gm[c];
    tbuf[tid] = 1.0f / (1.0f + expf(-s));
}

// ---- K4: generated kernels, stored transposed kt[b][g][cin][cout] -------
__global__ void __launch_bounds__(256)
k4_kernels(const float* __restrict__ tbuf, const float* __restrict__ Wk2,
           const float* __restrict__ bk2, float* __restrict__ kt) {
    int idx = blockIdx.x * 256 + threadIdx.x;     // [0, B*r*C*C)
    int b   = idx >> 15;                          // r*C*C = 32768
    int rem = idx & 32767;
    int g   = rem >> 12;                          // C*C = 4096
    int oc  = rem & 4095;
    int o   = oc >> 6;
    int cin = oc & 63;
    float s = bk2[g * 4096 + oc];
    const float* w  = Wk2 + ((size_t)g * 4096 + oc) * R_;
    const float* tg = tbuf + b * 64 + g * R_;
#pragma unroll
    for (int i = 0; i < R_; ++i) s += tg[i] * w[i];
    kt[(((size_t)b * R_ + g) * C_ + cin) * C_ + o] = s;
}

// ---- K5: per-bin GEMM via chained V_WMMA_F32_16X16X4_F32 ----------------
// grid = (512, 32), 256 threads = 8 waves; wave w handles tile blockIdx.x*8+w
// (16 binned pixels). Kernel matrix (16 KB) staged in LDS per workgroup.
__global__ void __launch_bounds__(256)
k5_wmma(const float* __restrict__ x1t, const float* __restrict__ kt,
        const int* __restrict__ counts, const int* __restrict__ binned,
        float* __restrict__ out) {
    __shared__ float sk[C_ * C_];                 // kt[bin]: [cin][cout]

    int bin   = blockIdx.y;
    int b     = bin >> 3;
    int count = counts[bin];
    if (blockIdx.x * 8 * 16 >= count) return;     // uniform exit (pre-barrier)

    const float* ksrc = kt + (size_t)bin * (C_ * C_);
    for (int i = threadIdx.x; i < C_ * C_; i += 256) sk[i] = ksrc[i];
    __syncthreads();

    int wave = threadIdx.x >> 5;
    int lane = threadIdx.x & 31;
    int tbase = (blockIdx.x * 8 + wave) * 16;
    if (tbase >= count) return;                   // wave-uniform exit

    int half = lane >> 4;                         // K half-select
    int n    = lane & 15;                         // row-for-A / col-for-B,D

    const int* binp = binned + (size_t)bin * HW_;

    // ---- A fragments: row M=n, K-pair (kk*4 + 2*half, +1). 16 K-steps.
    int am = tbase + n;
    int pa = binp[am < count ? am : tbase];       // clamp ragged tile
    const float* xrow = x1t + ((size_t)b * HW_ + pa) * C_;
    v2f afrag[16];
#pragma unroll
    for (int kk = 0; kk < 16; ++kk)
        afrag[kk] = *reinterpret_cast<const v2f*>(xrow + kk * 4 + 2 * half);

    // ---- D-row pixel ids: lane covers rows v + 8*half, v = 0..7
    int pd[8]; int rv[8];
#pragma unroll
    for (int v = 0; v < 8; ++v) {
        int row = tbase + v + 8 * half;
        rv[v] = (row < count);
        pd[v] = binp[rv[v] ? row : tbase];
    }

    float* outb = out + (size_t)b * C_ * HW_;
#pragma unroll
    for (int nc = 0; nc < 4; ++nc) {              // 4 x 16 output columns
        v8f acc = {};
#pragma unroll
        for (int kk = 0; kk < 16; ++kk) {         // K = 64 in steps of 4
            int c0 = kk * 4 + 2 * half;
            v2f bfrag;
            bfrag.x = sk[(c0    ) * C_ + nc * 16 + n];
            bfrag.y = sk[(c0 + 1) * C_ + nc * 16 + n];
            acc = __builtin_amdgcn_wmma_f32_16x16x4_f32(
                false, afrag[kk], false, bfrag, (short)0, acc, false, false);
        }
        // scatter rows (out is channel-major; binned pixels are scattered)
#pragma unroll
        for (int v = 0; v < 8; ++v) {
            if (rv[v]) outb[(size_t)(nc * 16 + n) * HW_ + pd[v]] = acc[v];
        }
    }
}

// ---------------------------------------------------------------------------
extern "C" void kernel_launch(void* const* d_in, const int* in_sizes, int n_in,
                              void* d_out, int out_size, void* d_ws, size_t ws_size,
                              hipStream_t stream) {
    const float* rgb  = (const float*)d_in[0];
    const float* edge = (const float*)d_in[1];
    const float* W1   = (const float*)d_in[2];
    const float* b1   = (const float*)d_in[3];
    const float* Wk1  = (const float*)d_in[4];
    const float* bk1  = (const float*)d_in[5];
    const float* Wk2  = (const float*)d_in[6];
    const float* bk2  = (const float*)d_in[7];
    const float* Wg   = (const float*)d_in[8];
    const float* bg   = (const float*)d_in[9];
    float* out = (float*)d_out;

    // workspace carve-up (all offsets 128B aligned); total ~76 MB
    char* ws = (char*)d_ws;
    float* x1t    = (float*)(ws);                         // 67,108,864 B
    float* gmean  = (float*)(ws + 67108864);              //      1,024 B
    float* tbuf   = (float*)(ws + 67109888);              //      1,024 B
    float* kt     = (float*)(ws + 67110912);              //    524,288 B
    int*   counts = (int*)  (ws + 67635200);              //        128 B
    int*   binned = (int*)  (ws + 67635328);              //  8,388,608 B

    k0_zero_counts<<<1, 32, 0, stream>>>(counts);
    k1_x1_guide_bin<<<(B_ * HW_) / 256, 256, 0, stream>>>(
        rgb, edge, W1, b1, Wg, bg, x1t, counts, binned);
    k2_mean<<<B_ * C_, 256, 0, stream>>>(x1t, gmean);
    k3_sigmoid<<<1, 256, 0, stream>>>(gmean, Wk1, bk1, tbuf);
    k4_kernels<<<(B_ * R_ * C_ * C_) / 256, 256, 0, stream>>>(tbuf, Wk2, bk2, kt);
    k5_wmma<<<dim3(512, B_ * R_), 256, 0, stream>>>(x1t, kt, counts, binned, out);
}